// WeightedGCN_18537078850139
// MI455X (gfx1250) — compile-verified
//
#include <hip/hip_runtime.h>
#include <math.h>

#define N_NODES   80000
#define N_EDGES   1280000
#define FEATS     64
#define ALPHA     0.8f
#define ROWS_PER_BLOCK 128
#define SA_STRIDE 65   // pad 64->65 to avoid stride-64 LDS bank conflicts

typedef __attribute__((ext_vector_type(2))) float v2f;
typedef __attribute__((ext_vector_type(8))) float v8f;

// ---------------------------------------------------------------------------
// Kernel 1: zero the scatter accumulators in d_ws (sum_m then sum_w)
// ---------------------------------------------------------------------------
__global__ void WGCN_zero_ws(float* __restrict__ ws, int n) {
    int i = blockIdx.x * blockDim.x + threadIdx.x;
    if (i < n) ws[i] = 0.0f;
}

// ---------------------------------------------------------------------------
// Kernel 2: edge gather + scatter-add.
// 16 threads per edge; each thread handles a float4 slice of the 64 features.
// Gathers hit L2 (feats = 20.5 MB << 192 MB L2); scatter via f32 atomics at L2.
// ---------------------------------------------------------------------------
__global__ __launch_bounds__(256) void WGCN_edge_scatter(
    const float* __restrict__ feats, const int* __restrict__ src,
    const int* __restrict__ dst, const float* __restrict__ ew,
    float* __restrict__ sum_m, float* __restrict__ sum_w) {
    long tid = (long)blockIdx.x * blockDim.x + threadIdx.x;
    if (tid >= (long)N_EDGES * 16) return;
    int e = (int)(tid >> 4);
    int c = ((int)tid & 15) * 4;
    int s = src[e];
    int d = dst[e];
    float w = ew[e];
    float4 f = *(const float4*)(feats + (long)s * FEATS + c);
    float* mp = sum_m + (long)d * FEATS + c;
    atomicAdd(mp + 0, w * f.x);
    atomicAdd(mp + 1, w * f.y);
    atomicAdd(mp + 2, w * f.z);
    atomicAdd(mp + 3, w * f.w);
    if ((tid & 15) == 0) atomicAdd(sum_w + d, w);
}

// ---------------------------------------------------------------------------
// Kernel 3: weighted-mean blend + [128x64]@[64x64] GEMM via V_WMMA_F32_16X16X4_F32
// 256 threads = 8 waves; each wave owns 16 node rows, computes all 64 output
// columns as 4 N-tiles of 16, chaining 16 K-steps (K=4) per tile.
// ---------------------------------------------------------------------------
__global__ __launch_bounds__(256) void WGCN_node_gemm(
    const float* __restrict__ feats, const float* __restrict__ sum_m,
    const float* __restrict__ sum_w, const float* __restrict__ W,
    const float* __restrict__ bias, float* __restrict__ out) {
    __shared__ float sA[ROWS_PER_BLOCK * SA_STRIDE];  // blended tile, padded
    __shared__ float sW[FEATS * FEATS];               // full weight matrix

    const int tid  = threadIdx.x;
    const int row0 = blockIdx.x * ROWS_PER_BLOCK;

    // Stage W (16 KB) into LDS, coalesced.
    for (int i = tid; i < FEATS * FEATS; i += 256) sW[i] = W[i];

    // aggregated = alpha*feats + (1-alpha)*(sum_w>0 ? sum_m/max(sum_w,1e-30) : 0)
    for (int i = tid; i < ROWS_PER_BLOCK * FEATS; i += 256) {
        int r = i >> 6;
        int c = i & 63;
        long g = (long)(row0 + r);
        float f  = feats[g * FEATS + c];
        float sm = sum_m[g * FEATS + c];
        float sw = sum_w[g];
        float hn = (sw > 0.0f) ? (sm / fmaxf(sw, 1e-30f)) : 0.0f;
        sA[r * SA_STRIDE + c] = ALPHA * f + (1.0f - ALPHA) * hn;
    }
    __syncthreads();

    const int wave = tid >> 5;
    const int lane = tid & 31;
    const int m    = lane & 15;   // row within 16-row tile (A), or N within tile (B/D)
    const int kh   = lane >> 4;   // K-half selector for A/B fragments
    const int rw   = wave * 16;   // this wave's local row base

    // A fragments for all 16 K-blocks: lane holds A[m][k0], A[m][k0+1],
    // k0 = 4*kb + 2*kh  (ISA 32-bit 16x4 A layout).
    v2f a[16];
#pragma unroll
    for (int kb = 0; kb < 16; ++kb) {
        const float* p = &sA[(rw + m) * SA_STRIDE + 4 * kb + 2 * kh];
        a[kb].x = p[0];
        a[kb].y = p[1];
    }

#pragma unroll
    for (int nt = 0; nt < 4; ++nt) {
        v8f acc = {};
#pragma unroll
        for (int kb = 0; kb < 16; ++kb) {
            const int k0 = 4 * kb + 2 * kh;
            v2f bf;
            bf.x = sW[(k0 + 0) * FEATS + nt * 16 + m];
            bf.y = sW[(k0 + 1) * FEATS + nt * 16 + m];
            // D = A x B + C   (f32, 16x16x4)
            acc = __builtin_amdgcn_wmma_f32_16x16x4_f32(
                /*neg_a=*/false, a[kb], /*neg_b=*/false, bf,
                /*c_mod=*/(short)0, acc, /*reuse_a=*/false, /*reuse_b=*/false);
        }
        // Epilogue: D VGPR r -> (M = r + 8*kh, N = m); add bias, sigmoid, store.
        const int col = nt * 16 + m;
        const float bb = bias[col];
#pragma unroll
        for (int r = 0; r < 8; ++r) {
            long grow = (long)(row0 + rw + r + (kh ? 8 : 0));
            float v = acc[r] + bb;
            out[grow * FEATS + col] = 1.0f / (1.0f + __expf(-v));
        }
    }
}

// ---------------------------------------------------------------------------
extern "C" void kernel_launch(void* const* d_in, const int* in_sizes, int n_in,
                              void* d_out, int out_size, void* d_ws, size_t ws_size,
                              hipStream_t stream) {
    const float* feats = (const float*)d_in[0];
    const int*   src   = (const int*)d_in[1];
    const int*   dst   = (const int*)d_in[2];
    const float* ew    = (const float*)d_in[3];
    const float* W     = (const float*)d_in[4];
    const float* bias  = (const float*)d_in[5];
    float*       out   = (float*)d_out;

    float* sum_m = (float*)d_ws;                       // [N_NODES * FEATS]
    float* sum_w = sum_m + (size_t)N_NODES * FEATS;    // [N_NODES]

    // 1) zero accumulators (deterministic per call)
    int nz = N_NODES * (FEATS + 1);
    WGCN_zero_ws<<<(nz + 255) / 256, 256, 0, stream>>>(sum_m, nz);

    // 2) edge scatter: 16 threads/edge
    long eth = (long)N_EDGES * 16;
    int eblocks = (int)((eth + 255) / 256);
    WGCN_edge_scatter<<<eblocks, 256, 0, stream>>>(feats, src, dst, ew, sum_m, sum_w);

    // 3) blend + WMMA GEMM + sigmoid  (80000 = 625 * 128, no tail)
    WGCN_node_gemm<<<N_NODES / ROWS_PER_BLOCK, 256, 0, stream>>>(
        feats, sum_m, sum_w, W, bias, out);
}